// GCN_72292889526468
// MI455X (gfx1250) — compile-verified
//
#include <hip/hip_runtime.h>

#define N_NODES 65536
#define N_EDGES 524288
#define N_SLI   512

typedef __attribute__((ext_vector_type(2))) float v2f;
typedef __attribute__((ext_vector_type(8))) float v8f;

// pointer types for the async global->LDS builtin (param types per hipcc diagnostic:
// AS1 "int __vector(4)*" source, AS3 dest)
typedef int v4i_ls __attribute__((vector_size(16)));
typedef __attribute__((address_space(1))) v4i_ls* as1_v4i;
typedef __attribute__((address_space(3))) v4i_ls* as3_v4i;

// ---------------- degree / norm ----------------
__global__ void deg_kernel(const int* __restrict__ src, const int* __restrict__ dst,
                           float* __restrict__ outdeg, float* __restrict__ indeg) {
    int e = blockIdx.x * blockDim.x + threadIdx.x;
    if (e < N_EDGES) {
        atomicAdd(&outdeg[src[e]], 1.0f);
        atomicAdd(&indeg[dst[e]], 1.0f);
    }
}

__global__ void rsqrt_kernel(float* __restrict__ d, int n) {
    int i = blockIdx.x * blockDim.x + threadIdx.x;
    if (i < n) d[i] = rsqrtf(fmaxf(d[i], 1.0f));
}

// ---------------- x [S,C,N] -> xT [S*N, C] ----------------
__global__ void transpose_kernel(const float* __restrict__ x, float* __restrict__ xT) {
    int idx = blockIdx.x * blockDim.x + threadIdx.x;   // over N_NODES*128
    int row = idx >> 7, c = idx & 127;
    int s = row >> 7, n = row & 127;
    xT[idx] = x[(s * 128 + c) * 128 + n];
}

// ------- edge scatter: agg[dst] += h[src] * rout[src], 4 channels/lane -------
__global__ void scatter_kernel(const float* __restrict__ h, int ldh,
                               const float* __restrict__ rout,
                               const int* __restrict__ src, const int* __restrict__ dst,
                               float* __restrict__ agg) {
    int idx = blockIdx.x * blockDim.x + threadIdx.x;   // over N_EDGES*32
    int e = idx >> 5, q = (idx & 31) << 2;
    int s = src[e], d = dst[e];
    float r = rout[s];
    float4 hv = *(const float4*)(h + s * ldh + q);
    float* ap = agg + d * 128 + q;
    atomicAdd(ap + 0, hv.x * r);
    atomicAdd(ap + 1, hv.y * r);
    atomicAdd(ap + 2, hv.z * r);
    atomicAdd(ap + 3, hv.w * r);
}

// ---------------- WMMA f32 GEMM: out = act(A(*rowscale) @ W + b) (+resid) ----------------
// A: [65536, K] stride lda.  WOK: W stored [O,K] row-major, else [K,O] row-major.
// P1: A-column k<256 comes from per-slice gfeat[256], k>=256 from A at (k-256).
// STAGE: W ([K][O], <=64KB) staged into LDS via async global->LDS; B read via ds_load.
// PRE:   A strip (16 x 128) preloaded + pre-scaled into VGPRs (k-loop fully unrolled
//        so areg[] indexing is constant -> stays in registers, no scratch).
template<int K, int O, bool WOK, bool LEAKY, bool RES, bool SCALE, bool P1>
__global__ __launch_bounds__(256)
void gemm_wmma(const float* __restrict__ A, int lda,
               const float* __restrict__ W,
               const float* __restrict__ bias,
               const float* __restrict__ rowscale,
               const float* __restrict__ resid, int ldr,
               const float* __restrict__ gfeat,
               float* __restrict__ out, int ldo)
{
    constexpr bool STAGE = (!WOK) && (K * O <= 16384);   // 64KB of f32 fits in LDS
    constexpr bool PRE   = (K == 128) && !P1;
    __shared__ float wlds[STAGE ? K * O : 1];

    if (STAGE) {
#if defined(__AMDGCN__) && __has_builtin(__builtin_amdgcn_global_load_async_to_lds_b128) && __has_builtin(__builtin_amdgcn_s_wait_asynccnt)
        for (int i = threadIdx.x * 4; i < K * O; i += 256 * 4)
            __builtin_amdgcn_global_load_async_to_lds_b128(
                (as1_v4i)(W + i), (as3_v4i)(wlds + i), 0, 0);
        __builtin_amdgcn_s_wait_asynccnt(0);
#else
        for (int i = threadIdx.x; i < K * O; i += 256) wlds[i] = W[i];
#endif
        __syncthreads();
    }

    const int lane = threadIdx.x & 31;
    const int wave = threadIdx.x >> 5;
    const int half = lane >> 4;      // 0: K pair {k,k+1}, 1: {k+2,k+3}
    const int l15  = lane & 15;
    const int rowTile = (blockIdx.x * 8 + wave) * 16;
    const int arow = rowTile + l15;  // A-matrix row for this lane (both halves cover M=0..15)
    const float rs = SCALE ? rowscale[arow] : 1.0f;
    const int slice = arow >> 7;

    v2f areg[PRE ? K / 4 : 1];
    if (PRE) {
#pragma unroll
        for (int k = 0; k < K; k += 4) {
            const int kb = k + half * 2;
            v2f av = *(const v2f*)(A + arow * lda + kb);
            if (SCALE) { av.x *= rs; av.y *= rs; }
            areg[k / 4] = av;
        }
    }

    for (int nt = 0; nt < O / 16; ++nt) {
        const int ccol = nt * 16 + l15;
        v8f acc = {};
        if (PRE) {
            // fully unrolled: areg[] indices are compile-time constants -> VGPRs
#pragma unroll
            for (int k = 0; k < K; k += 4) {
                const int kb = k + half * 2;
                v2f b;
                if (STAGE) {
                    b.x = wlds[kb * O + ccol];
                    b.y = wlds[(kb + 1) * O + ccol];
                } else {
                    b.x = W[kb * O + ccol];
                    b.y = W[(kb + 1) * O + ccol];
                }
                acc = __builtin_amdgcn_wmma_f32_16x16x4_f32(
                    false, areg[k / 4], false, b, (short)0, acc, false, false);
            }
        } else {
#pragma unroll 8
            for (int k = 0; k < K; k += 4) {
                const int kb = k + half * 2;
                v2f a, b;
                if (P1 && kb < 256)      a = *(const v2f*)(gfeat + slice * 256 + kb);
                else if (P1)             a = *(const v2f*)(A + arow * lda + (kb - 256));
                else                     a = *(const v2f*)(A + arow * lda + kb);
                if (SCALE) { a.x *= rs; a.y *= rs; }
                if (STAGE) {
                    b.x = wlds[kb * O + ccol];
                    b.y = wlds[(kb + 1) * O + ccol];
                } else if (WOK) {        // W[o][k], contiguous in k
                    b = *(const v2f*)(W + ccol * K + kb);
                } else {                 // W[k][o]
                    b.x = W[kb * O + ccol];
                    b.y = W[(kb + 1) * O + ccol];
                }
                acc = __builtin_amdgcn_wmma_f32_16x16x4_f32(
                    false, a, false, b, (short)0, acc, false, false);
            }
        }
        const float bv = bias[ccol];
#pragma unroll
        for (int r = 0; r < 8; ++r) {
            const int crow = rowTile + r + half * 8;   // C/D layout: VGPR r -> M=r / M=r+8
            float v = acc[r] + bv;
            if (LEAKY) v = (v >= 0.0f) ? v : 0.01f * v;
            if (RES)   v += resid[crow * ldr + ccol];
            out[crow * ldo + ccol] = v;
        }
    }
}

// ---------------- per-slice max over 128 nodes ----------------
__global__ void max_kernel(const float* __restrict__ fused, float* __restrict__ gmax) {
    int s = blockIdx.x, o = threadIdx.x;   // blockDim = 256
    float m = -3.402823466e38f;
    for (int n = 0; n < 128; ++n)
        m = fmaxf(m, fused[((s << 7) + n) * 256 + o]);
    gmax[s * 256 + o] = m;
}

// ---------------- final 64 -> 2 projection into [S,2,N] ----------------
__global__ void p3_kernel(const float* __restrict__ y2, const float* __restrict__ W,
                          const float* __restrict__ b, float* __restrict__ out) {
    int idx = blockIdx.x * blockDim.x + threadIdx.x;   // over N_NODES*2
    int row = idx >> 1, o = idx & 1;
    float s = b[o];
#pragma unroll
    for (int c = 0; c < 64; ++c) s += y2[row * 64 + c] * W[o * 64 + c];
    int sl = row >> 7, n = row & 127;
    out[(sl * 2 + o) * 128 + n] = s;
}

extern "C" void kernel_launch(void* const* d_in, const int* in_sizes, int n_in,
                              void* d_out, int out_size, void* d_ws, size_t ws_size,
                              hipStream_t stream) {
    const float* x        = (const float*)d_in[0];
    const int*   src      = (const int*)d_in[1];
    const int*   dst      = (const int*)d_in[2];
    const float* W_head   = (const float*)d_in[3];
    const float* b_head   = (const float*)d_in[4];
    const float* W_res[3] = {(const float*)d_in[5], (const float*)d_in[7], (const float*)d_in[9]};
    const float* b_res[3] = {(const float*)d_in[6], (const float*)d_in[8], (const float*)d_in[10]};
    const float* W_fusion = (const float*)d_in[11];
    const float* b_fusion = (const float*)d_in[12];
    const float* W_p1     = (const float*)d_in[13];
    const float* b_p1     = (const float*)d_in[14];
    const float* W_p2     = (const float*)d_in[15];
    const float* b_p2     = (const float*)d_in[16];
    const float* W_p3     = (const float*)d_in[17];
    const float* b_p3     = (const float*)d_in[18];

    float* ws    = (float*)d_ws;
    float* rout  = ws;                                 // [65536]
    float* rin   = ws + N_NODES;                       // [65536]
    float* xT    = ws + 2 * N_NODES;                   // [65536,128]
    float* state = xT + (size_t)N_NODES * 128;         // [65536,512]
    float* agg   = state + (size_t)N_NODES * 512;      // [65536,128]
    float* fused = agg + (size_t)N_NODES * 128;        // [65536,256] (reused as y1)
    float* gmax  = fused + (size_t)N_NODES * 256;      // [512,256]
    float* y2    = xT;                                 // reuse (xT dead after layer 0)

    (void)hipMemsetAsync(rout, 0, 2 * N_NODES * sizeof(float), stream);
    deg_kernel<<<N_EDGES / 256, 256, 0, stream>>>(src, dst, rout, rin);
    rsqrt_kernel<<<(2 * N_NODES) / 256, 256, 0, stream>>>(rout, 2 * N_NODES);
    transpose_kernel<<<(N_NODES * 128) / 256, 256, 0, stream>>>(x, xT);

    for (int l = 0; l < 4; ++l) {
        (void)hipMemsetAsync(agg, 0, (size_t)N_NODES * 128 * sizeof(float), stream);
        const float* hin = (l == 0) ? xT : (state + (l - 1) * 128);
        int ldh = (l == 0) ? 128 : 512;
        scatter_kernel<<<(N_EDGES * 32) / 256, 256, 0, stream>>>(hin, ldh, rout, src, dst, agg);
        const float* Wl = (l == 0) ? W_head : W_res[l - 1];
        const float* bl = (l == 0) ? b_head : b_res[l - 1];
        if (l == 0)
            gemm_wmma<128, 128, false, true, false, true, false><<<512, 256, 0, stream>>>(
                agg, 128, Wl, bl, rin, nullptr, 0, nullptr, state, 512);
        else
            gemm_wmma<128, 128, false, true, true, true, false><<<512, 256, 0, stream>>>(
                agg, 128, Wl, bl, rin, state + (l - 1) * 128, 512, nullptr, state + l * 128, 512);
    }

    // fusion: [65536,512] @ W_fusion[256,512]^T -> fused [65536,256]
    gemm_wmma<512, 256, true, false, false, false, false><<<512, 256, 0, stream>>>(
        state, 512, W_fusion, b_fusion, nullptr, nullptr, 0, nullptr, fused, 256);
    max_kernel<<<N_SLI, 256, 0, stream>>>(fused, gmax);
    // p1: concat(gmax, state) [768] -> 256, leaky; y1 written over fused
    gemm_wmma<768, 256, true, true, false, false, true><<<512, 256, 0, stream>>>(
        state, 512, W_p1, b_p1, nullptr, nullptr, 0, gmax, fused, 256);
    // p2: 256 -> 64, leaky
    gemm_wmma<256, 64, true, true, false, false, false><<<512, 256, 0, stream>>>(
        fused, 256, W_p2, b_p2, nullptr, nullptr, 0, nullptr, y2, 64);
    // p3: 64 -> 2 into [S,2,N]
    p3_kernel<<<(N_NODES * 2) / 256, 256, 0, stream>>>(y2, W_p3, b_p3, (float*)d_out);
}